// Transducer_50689204027780
// MI455X (gfx1250) — compile-verified
//
#include <hip/hip_runtime.h>
#include <stdint.h>

// -------- CDNA5 TDM (tensor_load_to_lds) helpers --------
typedef uint32_t v4u __attribute__((ext_vector_type(4)));
typedef int32_t  v8i __attribute__((ext_vector_type(8)));
typedef int32_t  v4i __attribute__((ext_vector_type(4)));

static constexpr int Bn = 32, Sn = 512, Tn = 2048;
static constexpr int TILE_T = 256;          // t columns per block (1 per thread)
static constexpr int TILE_S = 8;            // s rows per TDM stage (8KB/array)
static constexpr int NSTAGE = Sn / TILE_S;  // 64 stages cover all of S
static constexpr int NBUF   = 4;            // 4-deep TDM pipeline

// Issue one 2-D TDM load: tile_h rows x tile_w fp32, row stride in elements.
__device__ __forceinline__ void tdm_load_2d(uint32_t lds_off, const void* gptr,
                                            uint32_t tile_w, uint32_t tile_h,
                                            uint64_t row_stride_elems) {
  uint64_t ga = (uint64_t)gptr;
  v4u g0;
  g0.x = 1u;                                    // count=1, user mode, no gather
  g0.y = lds_off;                               // lds_addr [63:32]
  g0.z = (uint32_t)(ga & 0xFFFFFFFFu);          // global_addr[31:0]
  g0.w = (uint32_t)((ga >> 32) & 0x1FFFFFFu)    // global_addr[56:32]
         | (2u << 30);                          // type = 2 ("image")
  v8i g1;
  g1[0] = (int)(2u << 16);                      // wg_mask=0, data_size=2 (4B)
  g1[1] = (int)((tile_w & 0xFFFFu) << 16);      // tensor_dim0[15:0] @ [63:48]
  g1[2] = (int)(((tile_w >> 16) & 0xFFFFu) | ((tile_h & 0xFFFFu) << 16)); // dim0 hi | dim1 lo
  g1[3] = (int)(((tile_h >> 16) & 0xFFFFu) | ((tile_w & 0xFFFFu) << 16)); // dim1 hi | tile_dim0
  g1[4] = (int)(tile_h & 0xFFFFu);              // tile_dim1 | tile_dim2=0
  g1[5] = (int)(uint32_t)(row_stride_elems & 0xFFFFFFFFu);       // dim0_stride lo
  g1[6] = (int)(uint32_t)((row_stride_elems >> 32) & 0xFFFFu);   // dim0_stride hi | dim1_stride lo
  g1[7] = 0;
  v4i z4 = {0, 0, 0, 0};
#if defined(__clang_major__) && (__clang_major__ >= 23)
  v8i z8 = {0, 0, 0, 0, 0, 0, 0, 0};
  __builtin_amdgcn_tensor_load_to_lds(g0, g1, z4, z4, z8, 0);
#else
  __builtin_amdgcn_tensor_load_to_lds(g0, g1, z4, z4, 0);
#endif
}

__global__ __launch_bounds__(TILE_T)
void transducer_init_ctx_kernel(const float* __restrict__ px,
                                const float* __restrict__ py,
                                const int* __restrict__ y,
                                int* __restrict__ out) {
  __shared__ float sm_px[NBUF][TILE_S * TILE_T];   // 4 x 8KB
  __shared__ float sm_py[NBUF][TILE_S * TILE_T];   // 4 x 8KB  (64KB total)

  const int tid = threadIdx.x;
  const int tilesPerB = Tn / TILE_T;               // 8
  const int b  = blockIdx.x / tilesPerB;
  const int t0 = (blockIdx.x % tilesPerB) * TILE_T;

  // Tile bases at s=0 for this (b, t-tile). px rows stride T+1, py rows stride T.
  const float* pxb = px + (size_t)b * Sn * (size_t)(Tn + 1) + t0;
  const float* pyb = py + (size_t)b * (Sn + 1) * (size_t)Tn + t0;

  // Scalar (SGPR) condition -> s_cbranch skips TDM entirely for waves != 0.
  // (TDM ignores EXEC, so an EXEC-masked guard would NOT prevent issue.)
  const bool leader = (__builtin_amdgcn_readfirstlane((int)(tid >> 5)) == 0);

  if (leader) {  // prologue: fill pipeline 3 stages deep (6 DMAs outstanding)
#pragma unroll
    for (int p = 0; p < NBUF - 1; ++p) {
      tdm_load_2d((uint32_t)(uintptr_t)&sm_px[p][0],
                  pxb + (size_t)p * TILE_S * (Tn + 1), TILE_T, TILE_S,
                  (uint64_t)(Tn + 1));
      tdm_load_2d((uint32_t)(uintptr_t)&sm_py[p][0],
                  pyb + (size_t)p * TILE_S * Tn, TILE_T, TILE_S,
                  (uint64_t)Tn);
    }
  }

  float best   = -__builtin_huge_valf();
  int   best_s = 0;

  for (int stage = 0; stage < NSTAGE; ++stage) {
    const int cur = stage & (NBUF - 1);
    if (leader) {
      const int nxt = stage + (NBUF - 1);
      if (nxt < NSTAGE) {
        // buf[nxt & 3] == buf[(stage-1) & 3]: consumed last iteration, safe.
        const int nb = nxt & (NBUF - 1);
        tdm_load_2d((uint32_t)(uintptr_t)&sm_px[nb][0],
                    pxb + (size_t)nxt * TILE_S * (Tn + 1), TILE_T, TILE_S,
                    (uint64_t)(Tn + 1));
        tdm_load_2d((uint32_t)(uintptr_t)&sm_py[nb][0],
                    pyb + (size_t)nxt * TILE_S * Tn, TILE_T, TILE_S,
                    (uint64_t)Tn);
      }
      // Pending stages beyond the current one (in-order completion):
      const int rem = NSTAGE - 1 - stage;
      if      (rem >= 3) __builtin_amdgcn_s_wait_tensorcnt(6);
      else if (rem == 2) __builtin_amdgcn_s_wait_tensorcnt(4);
      else if (rem == 1) __builtin_amdgcn_s_wait_tensorcnt(2);
      else               __builtin_amdgcn_s_wait_tensorcnt(0);
    }
    __syncthreads();                            // buffer[cur] ready for all waves

    const int s0 = stage * TILE_S;
#pragma unroll
    for (int r = 0; r < TILE_S; ++r) {
      // lane-stride-1 LDS reads: conflict-free
      float v = sm_px[cur][r * TILE_T + tid] + sm_py[cur][r * TILE_T + tid];
      if (v > best) { best = v; best_s = s0 + r; }   // strict > == argmax first-index
    }
    __syncthreads();                            // buffer[cur] consumed; safe to refill
  }

  // context k: state = best_s + k; sos_y[j] = (j < 2) ? blank : y[j-2]
  const int t  = t0 + tid;
  const int i0 = best_s;
  const int i1 = best_s + 1;
  const int c0 = (i0 < 2) ? 0 : y[(size_t)b * Sn + (i0 - 2)];
  const int c1 = (i1 < 2) ? 0 : y[(size_t)b * Sn + (i1 - 2)];
  const size_t o = ((size_t)b * Tn + t) * 2;
  out[o + 0] = c0;
  out[o + 1] = c1;
}

extern "C" void kernel_launch(void* const* d_in, const int* in_sizes, int n_in,
                              void* d_out, int out_size, void* d_ws, size_t ws_size,
                              hipStream_t stream) {
  (void)in_sizes; (void)n_in; (void)out_size; (void)d_ws; (void)ws_size;
  const float* px = (const float*)d_in[0];   // (B, S, T+1) fp32
  const float* py = (const float*)d_in[1];   // (B, S+1, T) fp32
  const int*   y  = (const int*)d_in[2];     // (B, S) int32
  int* out = (int*)d_out;                    // (B, T, 2) int32

  dim3 grid(Bn * (Tn / TILE_T));             // 256 blocks (8 per batch row)
  dim3 block(TILE_T);                        // 256 threads = 8 wave32
  hipLaunchKernelGGL(transducer_init_ctx_kernel, grid, block, 0, stream,
                     px, py, y, out);
}